// SpectralGCN_7275674600509
// MI455X (gfx1250) — compile-verified
//
#include <hip/hip_runtime.h>

// SpectralGCN (2-layer GCN, two graphs, shared weights) for gfx1250.
// Memory-bound: edge gather + f32 scatter-atomics dominate (~2.5 GB total).
// GEMM uses V_WMMA_F32_16X16X4_F32 to keep full fp32 precision (GEMM cost is
// negligible at 3.3 GFLOP, so no precision downgrade is warranted).

typedef float v2f __attribute__((ext_vector_type(2)));
typedef float v8f __attribute__((ext_vector_type(8)));

#define FD 64   // feature dim

// ---------- small elementwise kernels ----------
__global__ void k_fill(float* __restrict__ p, float v, int n) {
    int i = blockIdx.x * blockDim.x + threadIdx.x;
    if (i < n) p[i] = v;
}

__global__ void k_count_deg(const int* __restrict__ dst, float* deg, int E) {
    int e = blockIdx.x * blockDim.x + threadIdx.x;
    if (e < E) unsafeAtomicAdd(&deg[dst[e]], 1.0f);
}

__global__ void k_rsqrt_inplace(float* __restrict__ d, int n) {
    int i = blockIdx.x * blockDim.x + threadIdx.x;
    if (i < n) {
        float v = d[i];
        d[i] = (v > 0.0f) ? rsqrtf(v) : 0.0f;
    }
}

__global__ void k_edge_norm(const int* __restrict__ src, const int* __restrict__ dst,
                            const float* __restrict__ dinv, float* __restrict__ norm, int E) {
    int e = blockIdx.x * blockDim.x + threadIdx.x;
    if (e < E) norm[e] = dinv[src[e]] * dinv[dst[e]];
}

// ---------- WMMA GEMM: Y[N,64] = X[N,64] @ W[64,64], fp32 ----------
// One wave computes a 16-row x 64-col panel. A panel loaded once (32 VGPRs),
// reused across the 4 column tiles. Requires nrows % 16 == 0 (N = 100000 = 16*6250).
__global__ void k_gemm_wmma(const float* __restrict__ X, const float* __restrict__ W,
                            float* __restrict__ Y, int nrows) {
    int wave = (blockIdx.x * blockDim.x + threadIdx.x) >> 5;
    int lane = threadIdx.x & 31;
    int row0 = wave * 16;
    if (row0 >= nrows) return;          // wave-uniform exit; EXEC stays all-1s for WMMA

    int half = lane >> 4;               // 0: lanes 0-15, 1: lanes 16-31
    int l    = lane & 15;

    // A: 16x4 f32 tile per k-step. Lane layout (ISA 7.12.2): lanes 0-15 hold
    // K={0,1}, lanes 16-31 hold K={2,3} for row M = lane%16 -> contiguous pair.
    const float* xrow = X + (size_t)(row0 + l) * FD;
    v2f a[16];
#pragma unroll
    for (int ks = 0; ks < 16; ++ks) {
        a[ks] = *(const v2f*)(xrow + ks * 4 + half * 2);
    }

#pragma unroll
    for (int nt = 0; nt < 4; ++nt) {
        int col = nt * 16 + l;
        v8f c = {};
#pragma unroll
        for (int ks = 0; ks < 16; ++ks) {
            int k = ks * 4 + half * 2;
            v2f b;
            b.x = W[(size_t)k * FD + col];
            b.y = W[(size_t)(k + 1) * FD + col];
            c = __builtin_amdgcn_wmma_f32_16x16x4_f32(
                    /*neg_a=*/false, a[ks], /*neg_b=*/false, b,
                    /*c_mod=*/(short)0, c, /*reuse_a=*/false, /*reuse_b=*/false);
        }
        // D layout: VGPR r -> row = row0 + half*8 + r, col = nt*16 + l
#pragma unroll
        for (int r = 0; r < 8; ++r) {
            Y[(size_t)(row0 + half * 8 + r) * FD + col] = c[r];
        }
    }
}

// ---------- aggregation ----------
// Initialize accumulator with the self-loop contribution: out = xw * dinv[i]^2
__global__ void k_selfloop_init(const float* __restrict__ XW, const float* __restrict__ dinv,
                                float* __restrict__ out, int n) {
    int i = blockIdx.x * blockDim.x + threadIdx.x;
    if (i < n) {
        float di = dinv[i >> 6];
        out[i] = XW[i] * di * di;
    }
}

// Scatter-add messages over edges: out[dst] += xw[src] * norm[e]
// e is wave-uniform (64 feats = 2 waves/edge): src/dst/norm loads are uniform,
// gather and atomic accesses are coalesced 128B per wave.
__global__ void k_scatter(const float* __restrict__ XW, const int* __restrict__ src,
                          const int* __restrict__ dst, const float* __restrict__ norm,
                          float* out, int total /* = E*64 */) {
    int tid = blockIdx.x * blockDim.x + threadIdx.x;
    if (tid >= total) return;
    int e = tid >> 6;
    int d = tid & 63;
    float v = XW[(size_t)src[e] * FD + d] * norm[e];
    unsafeAtomicAdd(&out[(size_t)dst[e] * FD + d], v);
}

__global__ void k_bias_relu(float* __restrict__ out, const float* __restrict__ b, int n) {
    int i = blockIdx.x * blockDim.x + threadIdx.x;
    if (i < n) {
        float v = out[i] + b[i & 63];
        out[i] = v > 0.0f ? v : 0.0f;
    }
}

extern "C" void kernel_launch(void* const* d_in, const int* in_sizes, int n_in,
                              void* d_out, int out_size, void* d_ws, size_t ws_size,
                              hipStream_t stream) {
    const int N = in_sizes[0] / FD;
    const float* x[2]  = { (const float*)d_in[0], (const float*)d_in[1] };
    const int    Es[2] = { in_sizes[2] / 2, in_sizes[3] / 2 };
    const int*   ei[2] = { (const int*)d_in[2], (const int*)d_in[3] };
    const float* W1 = (const float*)d_in[4];
    const float* b1 = (const float*)d_in[5];
    const float* W2 = (const float*)d_in[6];
    const float* b2 = (const float*)d_in[7];
    float* out = (float*)d_out;

    // Workspace carve (reused sequentially across both graphs):
    //   dinv[N] | norm[Emax] | bufA[N*64] | bufB[N*64]   (~56 MB total)
    int Emax = Es[0] > Es[1] ? Es[0] : Es[1];
    float* ws   = (float*)d_ws;
    float* dinv = ws;
    float* norm = dinv + N;
    float* bufA = norm + Emax;
    float* bufB = bufA + (size_t)N * FD;

    const int BT = 256;
    const int nd = N * FD;
    const int gemmBlocks = (N / 16 + 7) / 8;   // 8 waves/block, one 16-row panel per wave

    for (int g = 0; g < 2; ++g) {
        const int E = Es[g];
        const int* src = ei[g];
        const int* dst = ei[g] + E;
        float* oseg = out + (size_t)g * N * FD;
        const int ed = E * FD;

        // GCN normalization (with self-loops): deg -> dinv -> per-edge norm
        k_fill        <<<(N + BT - 1) / BT, BT, 0, stream>>>(dinv, 1.0f, N);
        k_count_deg   <<<(E + BT - 1) / BT, BT, 0, stream>>>(dst, dinv, E);
        k_rsqrt_inplace<<<(N + BT - 1) / BT, BT, 0, stream>>>(dinv, N);
        k_edge_norm   <<<(E + BT - 1) / BT, BT, 0, stream>>>(src, dst, dinv, norm, E);

        // Layer 1: x -> bufA (GEMM) -> bufB (aggregate) -> relu in-place
        k_gemm_wmma    <<<gemmBlocks, 256, 0, stream>>>(x[g], W1, bufA, N);
        k_selfloop_init<<<(nd + BT - 1) / BT, BT, 0, stream>>>(bufA, dinv, bufB, nd);
        k_scatter      <<<(ed + BT - 1) / BT, BT, 0, stream>>>(bufA, src, dst, norm, bufB, ed);
        k_bias_relu    <<<(nd + BT - 1) / BT, BT, 0, stream>>>(bufB, b1, nd);

        // Layer 2: bufB -> bufA (GEMM) -> aggregate directly into d_out segment
        k_gemm_wmma    <<<gemmBlocks, 256, 0, stream>>>(bufB, W2, bufA, N);
        k_selfloop_init<<<(nd + BT - 1) / BT, BT, 0, stream>>>(bufA, dinv, oseg, nd);
        k_scatter      <<<(ed + BT - 1) / BT, BT, 0, stream>>>(bufA, src, dst, norm, oseg, ed);
        k_bias_relu    <<<(nd + BT - 1) / BT, BT, 0, stream>>>(oseg, b2, nd);
    }
}